// MultiheadAttention_66666482369001
// MI455X (gfx1250) — compile-verified
//
#include <hip/hip_runtime.h>

// ---------------------------------------------------------------------------
// MultiheadAttention (B=4, T=1024, E=1024, H=16, hd=64), fp32 end-to-end
// using CDNA5 V_WMMA_F32_16X16X4_F32 (full reference precision).
// ---------------------------------------------------------------------------

typedef __attribute__((ext_vector_type(2))) float v2f;
typedef __attribute__((ext_vector_type(8))) float v8f;

__device__ __forceinline__ v8f wmma_f32(v2f a, v2f b, v8f c) {
  // D(16x16,f32) = A(16x4,f32) * B(4x16,f32) + C
  return __builtin_amdgcn_wmma_f32_16x16x4_f32(
      /*neg_a=*/false, a, /*neg_b=*/false, b,
      /*c_mod=*/(short)0, c, /*reuse_a=*/false, /*reuse_b=*/false);
}

#define LDS_STRIDE 20   // floats; pad so 16-lane fragment reads are bank-conflict
                        // free (20*l mod 64 distinct for l=0..15) and rows stay
                        // 16B-aligned for float4 stores / 8B for v2f loads.

// ---------------------------------------------------------------------------
// Kernel 1 & 5: C[M,N] = A[M,K](lda) * Bt[N,K](ldb)^T + bias[N], ldc row stride
// Block: 128x128 tile, 256 thr (8 waves), wave = 64x32 (4x2 WMMA tiles), KC=16
// ---------------------------------------------------------------------------
__global__ __launch_bounds__(256)
void mha_gemm_xwt_bias(const float* __restrict__ A, const float* __restrict__ Bt,
                       const float* __restrict__ bias, float* __restrict__ C,
                       int M, int N, int K, int lda, int ldb, int ldc) {
  __shared__ __align__(16) float As[128 * LDS_STRIDE];
  __shared__ __align__(16) float Bs[128 * LDS_STRIDE];

  const int tid  = threadIdx.x;
  const int lane = tid & 31, wave = tid >> 5;
  const int l15  = lane & 15, lhi = lane >> 4;
  const int wm   = wave & 1;        // 2 row groups of 64
  const int wn   = wave >> 1;       // 4 col groups of 32
  const int m0   = blockIdx.y * 128;
  const int n0   = blockIdx.x * 128;

  const int lrow  = tid >> 1;       // 0..127
  const int lhalf = (tid & 1) * 8;  // 0 or 8

  v8f acc[4][2] = {};

  for (int kc = 0; kc < K; kc += 16) {
    __syncthreads();
    // stage A tile [128 x 16]
    {
      const float* ap = A + (size_t)(m0 + lrow) * lda + kc + lhalf;
      float4 a0 = *(const float4*)ap;
      float4 a1 = *(const float4*)(ap + 4);
      *(float4*)&As[lrow * LDS_STRIDE + lhalf]     = a0;
      *(float4*)&As[lrow * LDS_STRIDE + lhalf + 4] = a1;
    }
    // stage B tile [128 x 16] (N-major, K contiguous == W layout)
    {
      const float* bp = Bt + (size_t)(n0 + lrow) * ldb + kc + lhalf;
      float4 b0 = *(const float4*)bp;
      float4 b1 = *(const float4*)(bp + 4);
      *(float4*)&Bs[lrow * LDS_STRIDE + lhalf]     = b0;
      *(float4*)&Bs[lrow * LDS_STRIDE + lhalf + 4] = b1;
    }
    __syncthreads();

#pragma unroll
    for (int kk = 0; kk < 4; ++kk) {
      const int kb = kk * 4 + 2 * lhi;  // A: lanes16-31 take K+2..K+3
      v2f af[4], bf[2];
#pragma unroll
      for (int mt = 0; mt < 4; ++mt)
        af[mt] = *(const v2f*)&As[(wm * 64 + mt * 16 + l15) * LDS_STRIDE + kb];
#pragma unroll
      for (int nt = 0; nt < 2; ++nt)
        bf[nt] = *(const v2f*)&Bs[(wn * 32 + nt * 16 + l15) * LDS_STRIDE + kb];
#pragma unroll
      for (int mt = 0; mt < 4; ++mt)
#pragma unroll
        for (int nt = 0; nt < 2; ++nt)
          acc[mt][nt] = wmma_f32(af[mt], bf[nt], acc[mt][nt]);
    }
  }

  // epilogue: C rows r and r+8 are 16-lane contiguous per acc VGPR
#pragma unroll
  for (int mt = 0; mt < 4; ++mt) {
    const int r0 = m0 + wm * 64 + mt * 16 + lhi * 8;
#pragma unroll
    for (int nt = 0; nt < 2; ++nt) {
      const int c  = n0 + wn * 32 + nt * 16 + l15;
      const float bv = bias[c];
#pragma unroll
      for (int r = 0; r < 8; ++r)
        C[(size_t)(r0 + r) * ldc + c] = acc[mt][nt][r] + bv;
    }
  }
}

// ---------------------------------------------------------------------------
// Kernel 2: raw[b,t,h,s] = 0.125 * sum_d q[b,t,h,d] * k[b,s,h,d]
// q/k live in qkv [4096, 3072]; output row stride H*T = 16384
// ---------------------------------------------------------------------------
__global__ __launch_bounds__(256)
void mha_scores(const float* __restrict__ qkv, float* __restrict__ raw) {
  __shared__ __align__(16) float As[128 * LDS_STRIDE];
  __shared__ __align__(16) float Bs[128 * LDS_STRIDE];

  const int bh = blockIdx.z, b = bh >> 4, h = bh & 15;
  const size_t qbase = (size_t)b * 1024 * 3072 + (size_t)h * 64;
  const size_t kbase = qbase + 1024;
  const size_t cbase = (size_t)b * (1024u * 16u * 1024u) + (size_t)h * 1024;

  const int tid  = threadIdx.x;
  const int lane = tid & 31, wave = tid >> 5;
  const int l15  = lane & 15, lhi = lane >> 4;
  const int wm   = wave & 1, wn = wave >> 1;
  const int t0   = blockIdx.y * 128;
  const int s0   = blockIdx.x * 128;
  const int lrow = tid >> 1, lhalf = (tid & 1) * 8;

  v8f acc[4][2] = {};

  for (int kc = 0; kc < 64; kc += 16) {
    __syncthreads();
    {
      const float* ap = qkv + qbase + (size_t)(t0 + lrow) * 3072 + kc + lhalf;
      *(float4*)&As[lrow * LDS_STRIDE + lhalf]     = *(const float4*)ap;
      *(float4*)&As[lrow * LDS_STRIDE + lhalf + 4] = *(const float4*)(ap + 4);
    }
    {
      const float* bp = qkv + kbase + (size_t)(s0 + lrow) * 3072 + kc + lhalf;
      *(float4*)&Bs[lrow * LDS_STRIDE + lhalf]     = *(const float4*)bp;
      *(float4*)&Bs[lrow * LDS_STRIDE + lhalf + 4] = *(const float4*)(bp + 4);
    }
    __syncthreads();

#pragma unroll
    for (int kk = 0; kk < 4; ++kk) {
      const int kb = kk * 4 + 2 * lhi;
      v2f af[4], bf[2];
#pragma unroll
      for (int mt = 0; mt < 4; ++mt)
        af[mt] = *(const v2f*)&As[(wm * 64 + mt * 16 + l15) * LDS_STRIDE + kb];
#pragma unroll
      for (int nt = 0; nt < 2; ++nt)
        bf[nt] = *(const v2f*)&Bs[(wn * 32 + nt * 16 + l15) * LDS_STRIDE + kb];
#pragma unroll
      for (int mt = 0; mt < 4; ++mt)
#pragma unroll
        for (int nt = 0; nt < 2; ++nt)
          acc[mt][nt] = wmma_f32(af[mt], bf[nt], acc[mt][nt]);
    }
  }

  const float scale = 0.125f;  // 1/sqrt(64)
#pragma unroll
  for (int mt = 0; mt < 4; ++mt) {
    const int r0 = t0 + wm * 64 + mt * 16 + lhi * 8;
#pragma unroll
    for (int nt = 0; nt < 2; ++nt) {
      const int c = s0 + wn * 32 + nt * 16 + l15;
#pragma unroll
      for (int r = 0; r < 8; ++r)
        raw[cbase + (size_t)(r0 + r) * 16384 + c] = scale * acc[mt][nt][r];
    }
  }
}

// ---------------------------------------------------------------------------
// Kernel 3: per score row (contiguous 1024 floats in raw layout):
// rowmax and sum(exp(x - rowmax)). One wave per row.
// ---------------------------------------------------------------------------
__global__ __launch_bounds__(256)
void mha_softmax_stats(const float* __restrict__ raw,
                       float* __restrict__ smax, float* __restrict__ ssum) {
  const int row  = blockIdx.x * 8 + (threadIdx.x >> 5);
  const int lane = threadIdx.x & 31;
  const float4* p = (const float4*)(raw + (size_t)row * 1024);

  float m = -3.402823466e38f;
#pragma unroll
  for (int i = 0; i < 8; ++i) {
    float4 v = p[lane + i * 32];
    m = fmaxf(m, fmaxf(fmaxf(v.x, v.y), fmaxf(v.z, v.w)));
  }
#pragma unroll
  for (int off = 16; off; off >>= 1) m = fmaxf(m, __shfl_xor(m, off, 32));

  float s = 0.f;
#pragma unroll
  for (int i = 0; i < 8; ++i) {
    float4 v = p[lane + i * 32];
    s += __expf(v.x - m) + __expf(v.y - m) + __expf(v.z - m) + __expf(v.w - m);
  }
#pragma unroll
  for (int off = 16; off; off >>= 1) s += __shfl_xor(s, off, 32);

  if (lane == 0) { smax[row] = m; ssum[row] = s; }
}

// ---------------------------------------------------------------------------
// Kernel 4: ctx[b,t,h*64+d] = sum_s softmax(raw)[t,s] * v[b,s,h*64+d]
// P re-materialized on the fly while staging into LDS (no 268MB P matrix).
// Block: 128 t-rows x 64 d-cols per (b,h); 8 waves, wave = 32x32 (2x2 tiles).
// ---------------------------------------------------------------------------
__global__ __launch_bounds__(256)
void mha_av(const float* __restrict__ raw, const float* __restrict__ qkv,
            const float* __restrict__ smax, const float* __restrict__ ssum,
            float* __restrict__ ctx) {
  __shared__ __align__(16) float Ps[128 * LDS_STRIDE];
  __shared__ __align__(16) float Vs[64 * LDS_STRIDE];  // transposed: [d][s]

  const int bh = blockIdx.y, b = bh >> 4, h = bh & 15;
  const int t0 = blockIdx.x * 128;
  const size_t rawbase = (size_t)b * (1024u * 16u * 1024u) + (size_t)h * 1024;
  const size_t vbase   = (size_t)b * 1024 * 3072 + 2048 + (size_t)h * 64;

  const int tid  = threadIdx.x;
  const int lane = tid & 31, wave = tid >> 5;
  const int l15  = lane & 15, lhi = lane >> 4;
  const int wm   = wave & 3;   // 4 row groups of 32
  const int wn   = wave >> 2;  // 2 col groups of 32

  // staging roles
  const int prow = tid >> 1, phalf = (tid & 1) * 8;     // P: 128x16
  const int vrow = tid >> 4, vcol = (tid & 15) * 4;     // V: 16x64

  const int statrow = (b * 1024 + t0 + prow) * 16 + h;  // == (b*T+t)*H+h
  const float rmax = smax[statrow];
  const float rinv = 1.f / ssum[statrow];

  v8f acc[2][2] = {};

  for (int kc = 0; kc < 1024; kc += 16) {
    __syncthreads();
    // stage P tile with on-the-fly softmax
    {
      const float* pp = raw + rawbase + (size_t)(t0 + prow) * 16384 + kc + phalf;
      float4 p0 = *(const float4*)pp;
      float4 p1 = *(const float4*)(pp + 4);
      float* pd = &Ps[prow * LDS_STRIDE + phalf];
      pd[0] = __expf(p0.x - rmax) * rinv;
      pd[1] = __expf(p0.y - rmax) * rinv;
      pd[2] = __expf(p0.z - rmax) * rinv;
      pd[3] = __expf(p0.w - rmax) * rinv;
      pd[4] = __expf(p1.x - rmax) * rinv;
      pd[5] = __expf(p1.y - rmax) * rinv;
      pd[6] = __expf(p1.z - rmax) * rinv;
      pd[7] = __expf(p1.w - rmax) * rinv;
    }
    // stage V tile transposed so B fragments read K-contiguous pairs
    {
      const float* vp = qkv + vbase + (size_t)(kc + vrow) * 3072 + vcol;
      float4 vv = *(const float4*)vp;
      Vs[(vcol + 0) * LDS_STRIDE + vrow] = vv.x;
      Vs[(vcol + 1) * LDS_STRIDE + vrow] = vv.y;
      Vs[(vcol + 2) * LDS_STRIDE + vrow] = vv.z;
      Vs[(vcol + 3) * LDS_STRIDE + vrow] = vv.w;
    }
    __syncthreads();

#pragma unroll
    for (int kk = 0; kk < 4; ++kk) {
      const int kb = kk * 4 + 2 * lhi;
      v2f af[2], bf[2];
#pragma unroll
      for (int mt = 0; mt < 2; ++mt)
        af[mt] = *(const v2f*)&Ps[(wm * 32 + mt * 16 + l15) * LDS_STRIDE + kb];
#pragma unroll
      for (int nt = 0; nt < 2; ++nt)
        bf[nt] = *(const v2f*)&Vs[(wn * 32 + nt * 16 + l15) * LDS_STRIDE + kb];
#pragma unroll
      for (int mt = 0; mt < 2; ++mt)
#pragma unroll
        for (int nt = 0; nt < 2; ++nt)
          acc[mt][nt] = wmma_f32(af[mt], bf[nt], acc[mt][nt]);
    }
  }

#pragma unroll
  for (int mt = 0; mt < 2; ++mt) {
    const int r0 = t0 + wm * 32 + mt * 16 + lhi * 8;
#pragma unroll
    for (int nt = 0; nt < 2; ++nt) {
      const int c = wn * 32 + nt * 16 + l15;  // 0..63
#pragma unroll
      for (int r = 0; r < 8; ++r)
        ctx[(size_t)(b * 1024 + r0 + r) * 1024 + h * 64 + c] = acc[mt][nt][r];
    }
  }
}

// ---------------------------------------------------------------------------
extern "C" void kernel_launch(void* const* d_in, const int* in_sizes, int n_in,
                              void* d_out, int out_size, void* d_ws, size_t ws_size,
                              hipStream_t stream) {
  const float* x     = (const float*)d_in[0];  // [4,1024,1024]
  const float* w_in  = (const float*)d_in[1];  // [3072,1024]
  const float* b_in  = (const float*)d_in[2];  // [3072]
  const float* w_out = (const float*)d_in[3];  // [1024,1024]
  const float* b_out = (const float*)d_in[4];  // [1024]

  float* out = (float*)d_out;                        // [4,1024,1024]
  float* raw = out + (size_t)4096 * 1024;            // [4,1024,16,1024]

  float* qkv  = (float*)d_ws;                        // [4096,3072]  48MB
  float* ctx  = qkv + (size_t)4096 * 3072;           // [4096,1024]  16MB
  float* smax = ctx + (size_t)4096 * 1024;           // [65536]
  float* ssum = smax + 65536;                        // [65536]

  // 1) qkv = x @ W_in^T + b_in
  mha_gemm_xwt_bias<<<dim3(3072 / 128, 4096 / 128), 256, 0, stream>>>(
      x, w_in, b_in, qkv, 4096, 3072, 1024, 1024, 1024, 3072);

  // 2) raw[b,t,h,s] = scale * q . k   (written straight into d_out's 2nd slot)
  mha_scores<<<dim3(8, 8, 64), 256, 0, stream>>>(qkv, raw);

  // 3) per-row softmax stats
  mha_softmax_stats<<<dim3(65536 / 8), 256, 0, stream>>>(raw, smax, ssum);

  // 4) ctx = softmax(raw) @ v, in [B,T,E] layout
  mha_av<<<dim3(8, 64), 256, 0, stream>>>(raw, qkv, smax, ssum, ctx);

  // 5) out = ctx @ W_out^T + b_out
  mha_gemm_xwt_bias<<<dim3(1024 / 128, 4096 / 128), 256, 0, stream>>>(
      ctx, w_out, b_out, out, 4096, 1024, 1024, 1024, 1024, 1024);
}